// DGPool_35527969472862
// MI455X (gfx1250) — compile-verified
//
#include <hip/hip_runtime.h>
#include <hip/hip_bf16.h>
#include <math.h>

// ---------------------------------------------------------------------------
// DGPool for MI455X (gfx1250, wave32).
// Pipeline: ||v|| -> WMMA matvec scores -> mean/std -> sigmoid+hist ->
// suffix-scan buckets -> scatter keys -> per-bucket rank (exact top-k order,
// ties by lower index) -> coalesced gather, + pool_loss.
// ---------------------------------------------------------------------------

#define TB 256
#define NBUCKET 65536

typedef float v2f __attribute__((ext_vector_type(2)));
typedef float v8f __attribute__((ext_vector_type(8)));

// Monotone order key: larger float -> larger uint (total order incl. sign).
__device__ __forceinline__ unsigned orderKey(float f) {
    unsigned u = __float_as_uint(f);
    return (u >> 31) ? ~u : (u | 0x80000000u);
}

// ---------------------------------------------------------------------------
__global__ void k_zero(float* wf, unsigned* hist) {
    int i = blockIdx.x * blockDim.x + threadIdx.x;
    if (i < NBUCKET) hist[i] = 0u;
    if (i == 0) { wf[1] = 0.f; wf[2] = 0.f; wf[5] = 0.f; }
}

__global__ void k_vnorm(const float* __restrict__ v, int F, float* wf) {
    __shared__ float lds[TB];
    float s = 0.f;
    for (int i = threadIdx.x; i < F; i += TB) { float t = v[i]; s += t * t; }
    lds[threadIdx.x] = s; __syncthreads();
    for (int st = TB / 2; st > 0; st >>= 1) {
        if (threadIdx.x < st) lds[threadIdx.x] += lds[threadIdx.x + st];
        __syncthreads();
    }
    if (threadIdx.x == 0) wf[0] = sqrtf(lds[0]);
}

// ---------------------------------------------------------------------------
// scores[r] = (x[r,:] . v) / (||v|| + eps) via V_WMMA_F32_16X16X4_F32.
// One wave handles 16 rows. A (16x4 f32): lanes 0-15 carry K0,K1; lanes 16-31
// carry K2,K3 (ISA 7.12.2). B (4x16) has every column equal to the v chunk,
// laid out with the matching lane-half split. D column N=0 holds the scores:
// lane 0 VGPRs 0..7 -> rows 0..7, lane 16 VGPRs 0..7 -> rows 8..15.
// ---------------------------------------------------------------------------
__global__ void k_scores_wmma(const float* __restrict__ x,
                              const float* __restrict__ v,
                              const float* __restrict__ wf,
                              float* __restrict__ scores, int n, int F) {
    int wave = (blockIdx.x * blockDim.x + threadIdx.x) >> 5;
    int row0 = wave * 16;
    if (row0 + 16 > n) return;               // wave-uniform exit (EXEC all-1s for WMMA)
    int lane = threadIdx.x & 31;
    int half = lane >> 4;                    // 0: K pair {0,1}, 1: K pair {2,3}
    int mrow = lane & 15;
    const float* xr = x + (size_t)(row0 + mrow) * F + 2 * half;
    const float* vp = v + 2 * half;
    float inv = 1.0f / (wf[0] + 1e-8f);

    v8f c = {};
    for (int kk = 0; kk < F; kk += 4) {
        v2f a; a.x = xr[kk]; a.y = xr[kk + 1];
        v2f b; b.x = vp[kk]; b.y = vp[kk + 1];   // replicated across all 16 columns
        c = __builtin_amdgcn_wmma_f32_16x16x4_f32(
                false, a, false, b, (short)0, c, false, false);
    }
    if (mrow == 0) {                          // N == 0 column
        int rb = row0 + (half ? 8 : 0);
#pragma unroll
        for (int g = 0; g < 8; ++g) scores[rb + g] = c[g] * inv;
    }
}

// Scalar tail for n not divisible by 16 (unused for n=100000, kept general).
__global__ void k_scores_tail(const float* __restrict__ x, const float* __restrict__ v,
                              const float* __restrict__ wf, float* __restrict__ scores,
                              int n, int F, int start) {
    int i = start + blockIdx.x * blockDim.x + threadIdx.x;
    if (i >= n) return;
    float inv = 1.0f / (wf[0] + 1e-8f);
    const float* xr = x + (size_t)i * F;
    float s = 0.f;
    for (int j = 0; j < F; ++j) s += xr[j] * v[j];
    scores[i] = s * inv;
}

// ---------------------------------------------------------------------------
__global__ void k_stats(const float* __restrict__ scores, int n, float* wf) {
    __shared__ float l1[TB];
    __shared__ float l2[TB];
    int i = blockIdx.x * blockDim.x + threadIdx.x;
    float s = (i < n) ? scores[i] : 0.f;
    l1[threadIdx.x] = s; l2[threadIdx.x] = s * s; __syncthreads();
    for (int st = TB / 2; st > 0; st >>= 1) {
        if (threadIdx.x < st) {
            l1[threadIdx.x] += l1[threadIdx.x + st];
            l2[threadIdx.x] += l2[threadIdx.x + st];
        }
        __syncthreads();
    }
    if (threadIdx.x == 0) { atomicAdd(&wf[1], l1[0]); atomicAdd(&wf[2], l2[0]); }
}

__global__ void k_finalize(float* wf, int n) {
    float mean = wf[1] / (float)n;
    float var  = wf[2] / (float)n - mean * mean;
    if (var < 0.f) var = 0.f;
    wf[3] = mean;
    wf[4] = 1.0f / (sqrtf(var) + 1e-8f);
}

// sigmoid, pool_loss partial, 64K-bucket histogram of order-key high 16 bits
__global__ void k_sig(const float* __restrict__ scores, float* __restrict__ sigArr,
                      unsigned* __restrict__ hist, float* wf, int n) {
    __shared__ float l[TB];
    int i = blockIdx.x * blockDim.x + threadIdx.x;
    float p = 0.f;
    if (i < n) {
        float z = (scores[i] - wf[3]) * wf[4];
        float s = 1.0f / (1.0f + expf(-z));
        sigArr[i] = s;
        p = s * (1.0f - s);
        atomicAdd(&hist[orderKey(s) >> 16], 1u);
    }
    l[threadIdx.x] = p; __syncthreads();
    for (int st = TB / 2; st > 0; st >>= 1) {
        if (threadIdx.x < st) l[threadIdx.x] += l[threadIdx.x + st];
        __syncthreads();
    }
    if (threadIdx.x == 0) atomicAdd(&wf[5], l[0]);
}

// ---- two-level descending (suffix) scan of hist[65536] ----
__global__ void k_scan1(const unsigned* __restrict__ hist, unsigned* __restrict__ part) {
    __shared__ unsigned l[TB];
    int b = blockIdx.x * TB + threadIdx.x;
    l[threadIdx.x] = hist[b]; __syncthreads();
    for (int st = TB / 2; st > 0; st >>= 1) {
        if (threadIdx.x < st) l[threadIdx.x] += l[threadIdx.x + st];
        __syncthreads();
    }
    if (threadIdx.x == 0) part[blockIdx.x] = l[0];
}

__global__ void k_scan2(const unsigned* __restrict__ part, unsigned* __restrict__ csfx) {
    __shared__ unsigned l[TB];
    l[threadIdx.x] = part[threadIdx.x]; __syncthreads();
    unsigned s = 0;
    for (int j = threadIdx.x + 1; j < TB; ++j) s += l[j];
    csfx[threadIdx.x] = s;   // elements in chunks strictly above mine
}

__global__ void k_scan3(const unsigned* __restrict__ hist, const unsigned* __restrict__ csfx,
                        unsigned* __restrict__ base, unsigned* __restrict__ cursor) {
    __shared__ unsigned l[TB];
    int b = blockIdx.x * TB + threadIdx.x;
    l[threadIdx.x] = hist[b]; __syncthreads();
    unsigned s = 0;
    for (int j = threadIdx.x + 1; j < TB; ++j) s += l[j];
    unsigned v = csfx[blockIdx.x] + s;   // # elements with strictly greater bucket
    base[b] = v;
    cursor[b] = v;
}

// scatter 64-bit keys into their bucket region (descending order over buckets)
__global__ void k_scatter(const float* __restrict__ sigArr, unsigned* __restrict__ cursor,
                          unsigned long long* __restrict__ key, int n) {
    int i = blockIdx.x * blockDim.x + threadIdx.x;
    if (i >= n) return;
    unsigned kk = orderKey(sigArr[i]);
    unsigned slot = atomicAdd(&cursor[kk >> 16], 1u);
    // high: order key (descending sort); low: ~idx so lower index wins ties
    key[slot] = ((unsigned long long)kk << 32)
              | (unsigned long long)(0xFFFFFFFFu - (unsigned)i);
}

// exact rank within each bucket (avg occupancy ~1.5), write perm[rank]=idx
__global__ void k_rank(const unsigned* __restrict__ hist, const unsigned* __restrict__ base,
                       const unsigned long long* __restrict__ key,
                       unsigned* __restrict__ perm, int kSel) {
    int b = blockIdx.x * blockDim.x + threadIdx.x;
    if (b >= NBUCKET) return;
    unsigned m = hist[b];
    if (m == 0) return;
    unsigned st = base[b];
    for (unsigned e = 0; e < m; ++e) {
        unsigned long long ke = key[st + e];
        unsigned cnt = 0;
        for (unsigned j = 0; j < m; ++j) cnt += (key[st + j] > ke) ? 1u : 0u;
        unsigned r = st + cnt;
        if (r < (unsigned)kSel)
            perm[r] = 0xFFFFFFFFu - (unsigned)(ke & 0xFFFFFFFFull);
    }
}

// one block per output row: out[r,:] = x[idx,:] * sig[idx]  (coalesced)
__global__ void k_gather(const float* __restrict__ x, const float* __restrict__ sigArr,
                         const unsigned* __restrict__ perm, float* __restrict__ out, int F) {
    int r = blockIdx.x;
    unsigned idx = perm[r];
    const float* xr = x + (size_t)idx * F;
    __builtin_prefetch(xr, 0, 0);            // global_prefetch_b8
    float s = sigArr[idx];
    float* o = out + (size_t)r * F;
    for (int j = threadIdx.x; j < F; j += blockDim.x) o[j] = xr[j] * s;
}

__global__ void k_loss(const float* wf, float* __restrict__ out, int n, size_t pos) {
    if (threadIdx.x == 0 && blockIdx.x == 0) out[pos] = wf[5] / (float)n;
}

// ---------------------------------------------------------------------------
extern "C" void kernel_launch(void* const* d_in, const int* in_sizes, int n_in,
                              void* d_out, int out_size, void* d_ws, size_t ws_size,
                              hipStream_t stream) {
    const float* x = (const float*)d_in[0];   // [n, F]
    const float* v = (const float*)d_in[1];   // [F, 1]
    float* out = (float*)d_out;               // [k*F] new_x  ++  [1] pool_loss

    int F = in_sizes[1];
    int n = in_sizes[0] / F;
    int k = (int)(n * 0.5);                   // POOL_RATIO = 0.5
    if (k < 1) k = 1;

    // workspace layout
    char* ws = (char*)d_ws;
    size_t off = 0;
    float* wf = (float*)(ws + off);                    off += 64;           // scalars
    float* scores = (float*)(ws + off);                off += (size_t)n * 4;
    float* sigArr = (float*)(ws + off);                off += (size_t)n * 4;
    unsigned* hist   = (unsigned*)(ws + off);          off += (size_t)NBUCKET * 4;
    unsigned* base   = (unsigned*)(ws + off);          off += (size_t)NBUCKET * 4;
    unsigned* cursor = (unsigned*)(ws + off);          off += (size_t)NBUCKET * 4;
    unsigned* part   = (unsigned*)(ws + off);          off += 256 * 4;
    unsigned* csfx   = (unsigned*)(ws + off);          off += 256 * 4;
    off = (off + 7) & ~(size_t)7;
    unsigned long long* key = (unsigned long long*)(ws + off); off += (size_t)n * 8;
    unsigned* perm = (unsigned*)(ws + off);            off += (size_t)k * 4;
    (void)ws_size; (void)n_in; (void)out_size;

    int nBlocks = (n + TB - 1) / TB;

    // 0) zero accumulators + histogram
    k_zero<<<NBUCKET / TB, TB, 0, stream>>>(wf, hist);
    // 1) ||v||
    k_vnorm<<<1, TB, 0, stream>>>(v, F, wf);
    // 2) scores via WMMA (16 rows / wave), scalar tail for leftovers
    int fullTiles = n / 16;
    if (fullTiles > 0) {
        int waves = fullTiles;
        int blocks = (waves * 32 + TB - 1) / TB;
        k_scores_wmma<<<blocks, TB, 0, stream>>>(x, v, wf, scores, n, F);
    }
    int rem = n - fullTiles * 16;
    if (rem > 0)
        k_scores_tail<<<(rem + TB - 1) / TB, TB, 0, stream>>>(x, v, wf, scores, n, F,
                                                              fullTiles * 16);
    // 3) mean / std
    k_stats<<<nBlocks, TB, 0, stream>>>(scores, n, wf);
    k_finalize<<<1, 1, 0, stream>>>(wf, n);
    // 4) sigmoid + pool_loss + bucket histogram
    k_sig<<<nBlocks, TB, 0, stream>>>(scores, sigArr, hist, wf, n);
    // 5) descending suffix scan over 64K buckets
    k_scan1<<<NBUCKET / TB, TB, 0, stream>>>(hist, part);
    k_scan2<<<1, TB, 0, stream>>>(part, csfx);
    k_scan3<<<NBUCKET / TB, TB, 0, stream>>>(hist, csfx, base, cursor);
    // 6) scatter keys, exact per-bucket rank -> perm
    k_scatter<<<nBlocks, TB, 0, stream>>>(sigArr, cursor, key, n);
    k_rank<<<NBUCKET / TB, TB, 0, stream>>>(hist, base, key, perm, k);
    // 7) gather output rows + loss scalar
    k_gather<<<k, TB, 0, stream>>>(x, sigArr, perm, out, F);
    k_loss<<<1, 1, 0, stream>>>(wf, out, n, (size_t)k * F);
}